// MutualInformationLoss_14912126452371
// MI455X (gfx1250) — compile-verified
//
#include <hip/hip_runtime.h>
#include <math.h>

typedef __attribute__((ext_vector_type(2))) float v2f;
typedef __attribute__((ext_vector_type(8))) float v8f;

#define NB     4096                 // batch
#define DIM    1024                 // feature dim
#define INV_T  10.0f                // 1 / temperature
#define MT     128                  // macro tile (rows == cols)
#define KC     16                   // K chunk staged in LDS
#define NCHUNK (DIM / KC)           // 64
#define LDP    20                   // LDS row pitch in floats (16 + 4 pad: 16B aligned, bank-clean)
#define NCG    (NB / 64)            // 64 column groups of 64 columns
#define NBLK   ((NB / MT) * (NB / MT))   // 1024 blocks
#define NPOSP  (NBLK * 8)           // one positive partial per wave

// ---------------------------------------------------------------------------
// CDNA5 async global->LDS copy (ASYNCcnt-tracked). Inline asm: bypasses the
// builtin's unspellable __device__/AS-qualified pointer parameter types.
// VGLOBAL GV form: vdst = LDS byte offset VGPR, vaddr = 64-bit global address.
// ---------------------------------------------------------------------------
__device__ __forceinline__ void async_copy_b128(const float* g, float* l) {
    const unsigned           lds = (unsigned)(uintptr_t)l;       // low 32 bits = LDS offset
    const unsigned long long ga  = (unsigned long long)(uintptr_t)g;
    asm volatile("global_load_async_to_lds_b128 %0, %1, off"
                 :: "v"(lds), "v"(ga)
                 : "memory");
}

template <int N>
__device__ __forceinline__ void wait_asynccnt() {
#if __has_builtin(__builtin_amdgcn_s_wait_asynccnt)
    __builtin_amdgcn_s_wait_asynccnt((short)N);
#else
    asm volatile("s_wait_asynccnt %0" :: "i"(N) : "memory");
#endif
}

// ---------------------------------------------------------------------------
// K1: L2-normalize each row (one wave32 per row), eps clamp like F.normalize.
// ---------------------------------------------------------------------------
__global__ void __launch_bounds__(256)
k_normalize(const float* __restrict__ feat, float* __restrict__ fhat) {
    const int row  = (blockIdx.x * blockDim.x + threadIdx.x) >> 5;
    const int lane = threadIdx.x & 31;
    const float4* src = reinterpret_cast<const float4*>(feat + (size_t)row * DIM);
    float4*       dst = reinterpret_cast<float4*>(fhat + (size_t)row * DIM);

    float4 v[8];
    float ss = 0.0f;
#pragma unroll
    for (int c = 0; c < 8; ++c) {
        v[c] = src[lane + c * 32];
        ss += v[c].x * v[c].x + v[c].y * v[c].y + v[c].z * v[c].z + v[c].w * v[c].w;
    }
#pragma unroll
    for (int m = 1; m < 32; m <<= 1) ss += __shfl_xor(ss, m, 32);
    const float inv = 1.0f / fmaxf(sqrtf(ss), 1e-12f);
#pragma unroll
    for (int c = 0; c < 8; ++c) {
        float4 o = v[c];
        o.x *= inv; o.y *= inv; o.z *= inv; o.w *= inv;
        dst[lane + c * 32] = o;
    }
}

// ---------------------------------------------------------------------------
// K2: 128x128 macro-tile of sim = fhat @ fhat^T per 256-thread block.
// Double-buffered async-to-LDS panel staging, f32 WMMA from LDS, fused
// exp/mask epilogue with deterministic partial outputs (no float atomics).
//   rowPart[i * NCG + cg] = sum_{j in 64-col group cg, j != i} exp(10*sim_ij)
//   posPart[waveId]       = sum over wave tile of exp where labels match, i!=j
// ---------------------------------------------------------------------------
__global__ void __launch_bounds__(256)
k_simtiles(const float* __restrict__ fhat,
           const long long* __restrict__ tgt,
           float* __restrict__ rowPart,
           float* __restrict__ posPart) {
    __shared__ float sA[2][MT][LDP];
    __shared__ float sB[2][MT][LDP];

    const int tid  = threadIdx.x;
    const int wv   = tid >> 5;
    const int lane = tid & 31;
    const int half = lane >> 4;      // K-split half of the wave
    const int l16  = lane & 15;      // M (A) / N (B) index within a 16x16 tile

    const int bx = blockIdx.x & ((NB / MT) - 1);   // 0..31 col macro
    const int by = blockIdx.x >> 5;                // 0..31 row macro
    const int macroRow = by * MT;
    const int macroCol = bx * MT;

    const int waveRow = wv >> 1;                   // 0..3 -> 32-row slice
    const int waveCol = wv & 1;                    // 0..1 -> 64-col slice
    const int rb = macroRow + waveRow * 32;        // wave row base
    const int cb = macroCol + waveCol * 64;        // wave col base

    // LDS fragment rows for this lane
    const int aRow0 = waveRow * 32 + l16;
    const int aRow1 = aRow0 + 16;
    const int bRowBase = waveCol * 64 + l16;

    // Stage one KC-wide panel pair (A rows + B rows) with 4 async B128/thread.
    auto stage = [&](int k, int buf) {
#pragma unroll
        for (int rep = 0; rep < 2; ++rep) {
            const int idx = tid + rep * 256;       // 0..511
            const int row = idx >> 2;              // 0..127
            const int q   = (idx & 3) * 4;         // 0,4,8,12 (floats)
            async_copy_b128(fhat + (size_t)(macroRow + row) * DIM + k + q,
                            &sA[buf][row][q]);
            async_copy_b128(fhat + (size_t)(macroCol + row) * DIM + k + q,
                            &sB[buf][row][q]);
        }
    };

    v8f acc[2][4];
#pragma unroll
    for (int r = 0; r < 2; ++r)
#pragma unroll
        for (int t = 0; t < 4; ++t) acc[r][t] = (v8f){0, 0, 0, 0, 0, 0, 0, 0};

    stage(0, 0);
    for (int c = 0; c < NCHUNK; ++c) {
        const int buf = c & 1;
        if (c + 1 < NCHUNK) {
            stage((c + 1) * KC, buf ^ 1);
            wait_asynccnt<4>();   // 4 newest (next chunk) may remain in flight
        } else {
            wait_asynccnt<0>();
        }
        __syncthreads();          // panels for `buf` visible block-wide

#pragma unroll
        for (int kk = 0; kk < KC; kk += 4) {
            const int ko = kk + 2 * half;          // K = 2*half + vgpr
            v2f a0 = *reinterpret_cast<const v2f*>(&sA[buf][aRow0][ko]);
            v2f a1 = *reinterpret_cast<const v2f*>(&sA[buf][aRow1][ko]);
            v2f b0 = *reinterpret_cast<const v2f*>(&sB[buf][bRowBase +  0][ko]);
            v2f b1 = *reinterpret_cast<const v2f*>(&sB[buf][bRowBase + 16][ko]);
            v2f b2 = *reinterpret_cast<const v2f*>(&sB[buf][bRowBase + 32][ko]);
            v2f b3 = *reinterpret_cast<const v2f*>(&sB[buf][bRowBase + 48][ko]);
            acc[0][0] = __builtin_amdgcn_wmma_f32_16x16x4_f32(false, a0, false, b0, (short)0, acc[0][0], false, false);
            acc[0][1] = __builtin_amdgcn_wmma_f32_16x16x4_f32(false, a0, false, b1, (short)0, acc[0][1], false, false);
            acc[0][2] = __builtin_amdgcn_wmma_f32_16x16x4_f32(false, a0, false, b2, (short)0, acc[0][2], false, false);
            acc[0][3] = __builtin_amdgcn_wmma_f32_16x16x4_f32(false, a0, false, b3, (short)0, acc[0][3], false, false);
            acc[1][0] = __builtin_amdgcn_wmma_f32_16x16x4_f32(false, a1, false, b0, (short)0, acc[1][0], false, false);
            acc[1][1] = __builtin_amdgcn_wmma_f32_16x16x4_f32(false, a1, false, b1, (short)0, acc[1][1], false, false);
            acc[1][2] = __builtin_amdgcn_wmma_f32_16x16x4_f32(false, a1, false, b2, (short)0, acc[1][2], false, false);
            acc[1][3] = __builtin_amdgcn_wmma_f32_16x16x4_f32(false, a1, false, b3, (short)0, acc[1][3], false, false);
        }
        __syncthreads();          // all waves done reading `buf` before rewrite
    }

    // ---- fused epilogue -------------------------------------------------
    // C/D layout: lane L, vgpr v -> M = v + 8*(L/16), N = L%16
    int tr[2][8];
#pragma unroll
    for (int r = 0; r < 2; ++r)
#pragma unroll
        for (int v = 0; v < 8; ++v)
            tr[r][v] = (int)tgt[rb + r * 16 + v + 8 * half];

    float rAcc[2][8] = {};
    float pos = 0.0f;
#pragma unroll
    for (int t = 0; t < 4; ++t) {
        const int j  = cb + t * 16 + l16;
        const int tc = (int)tgt[j];
#pragma unroll
        for (int r = 0; r < 2; ++r)
#pragma unroll
            for (int v = 0; v < 8; ++v) {
                const int i   = rb + r * 16 + v + 8 * half;
                const float e = expf(acc[r][t][v] * INV_T);
                if (i != j) {
                    rAcc[r][v] += e;
                    if (tr[r][v] == tc) pos += e;
                }
            }
    }

    // Row sums: reduce across the 16 lanes of each half (xor 1,2,4,8 stays in-half).
    const int colGroup = bx * 2 + waveCol;         // 0..63
#pragma unroll
    for (int r = 0; r < 2; ++r)
#pragma unroll
        for (int v = 0; v < 8; ++v) {
            float s = rAcc[r][v];
#pragma unroll
            for (int m = 1; m <= 8; m <<= 1) s += __shfl_xor(s, m, 32);
            if (l16 == 0)
                rowPart[(size_t)(rb + r * 16 + v + 8 * half) * NCG + colGroup] = s;
        }

    // Positive sum: full-wave reduction, one deterministic slot per wave.
#pragma unroll
    for (int m = 1; m < 32; m <<= 1) pos += __shfl_xor(pos, m, 32);
    if (lane == 0) posPart[blockIdx.x * 8 + wv] = pos;
}

// ---------------------------------------------------------------------------
// K3: deterministic final reduction.
// loss = -log(sum pos) + (1/NB) * sum_i log(sum_g rowPart[i][g])
// ---------------------------------------------------------------------------
__global__ void __launch_bounds__(256)
k_finalize(const float* __restrict__ rowPart,
           const float* __restrict__ posPart,
           float* __restrict__ out) {
    __shared__ float sNeg[256];
    __shared__ float sPos[256];
    const int tid = threadIdx.x;

    float negAcc = 0.0f;
    for (int i = tid; i < NB; i += 256) {
        const float* rp = rowPart + (size_t)i * NCG;
        float s = 0.0f;
#pragma unroll
        for (int g = 0; g < NCG; ++g) s += rp[g];
        negAcc += logf(s);
    }
    float posAcc = 0.0f;
    for (int w = tid; w < NPOSP; w += 256) posAcc += posPart[w];

    sNeg[tid] = negAcc;
    sPos[tid] = posAcc;
    __syncthreads();
    for (int off = 128; off > 0; off >>= 1) {
        if (tid < off) {
            sNeg[tid] += sNeg[tid + off];
            sPos[tid] += sPos[tid + off];
        }
        __syncthreads();
    }
    if (tid == 0) out[0] = -logf(sPos[0]) + sNeg[0] * (1.0f / (float)NB);
}

// ---------------------------------------------------------------------------
extern "C" void kernel_launch(void* const* d_in, const int* in_sizes, int n_in,
                              void* d_out, int out_size, void* d_ws, size_t ws_size,
                              hipStream_t stream) {
    (void)in_sizes; (void)n_in; (void)out_size; (void)ws_size;

    const float*     feat = (const float*)d_in[0];
    const long long* tgt  = (const long long*)d_in[1];
    float*           out  = (float*)d_out;

    char* ws = (char*)d_ws;
    float* fhat    = (float*)ws;                                          // 16 MB
    float* rowPart = (float*)(ws + (size_t)NB * DIM * sizeof(float));     // 1 MB
    float* posPart = (float*)(ws + (size_t)NB * DIM * sizeof(float)
                                 + (size_t)NB * NCG * sizeof(float));     // 32 KB

    // K1: 4096 rows, 8 waves (rows) per 256-thread block.
    k_normalize<<<NB / 8, 256, 0, stream>>>(feat, fhat);
    // K2: 32x32 macro-tiles of 128x128, 8 waves per block.
    k_simtiles<<<NBLK, 256, 0, stream>>>(fhat, tgt, rowPart, posPart);
    // K3: single-block ordered reduction.
    k_finalize<<<1, 256, 0, stream>>>(rowPart, posPart, out);
}